// ChunkRanker_22978075034013
// MI455X (gfx1250) — compile-verified
//
#include <hip/hip_runtime.h>

// Use raw ext-vector types (HIP's float4 may be a struct; these are clang
// vectors, valid for __builtin_nontemporal_load and WMMA operands).
typedef float v2f __attribute__((ext_vector_type(2)));
typedef float v4f __attribute__((ext_vector_type(4)));
typedef float v8f __attribute__((ext_vector_type(8)));

#define TPB   256    // 8 wave32s per block
#define ELEMS 8192   // 128*64 floats per chunk
#define V4S   2048   // float4s per chunk
#define ITERS 8      // V4S / TPB
#define SV4   160    // float4s in the 640-elem "start" region (first 10 rows)
#define CV4   1888   // float4 offset of previous_context[-10:] (= (8192-640)/4)

__global__ __launch_bounds__(TPB) void chunkrank_kernel(
    const v4f* __restrict__ chunks,
    const v4f* __restrict__ prevctx,
    float* __restrict__ out)
{
    __shared__ float smem[5 * 8];   // 5 quantities x 8 waves

    const int n = blockIdx.x;
    const int t = threadIdx.x;
    const v4f* cp = chunks + (size_t)n * V4S;

    float s = 0.f, ss = 0.f, dt = 0.f, sts = 0.f, cs = 0.f;

    // Iteration 0: float4 indices [0,256) -> contains the entire 640-element
    // "start" region (first 160 float4s). Fuse the cosine terms in here.
    {
        v4f x = __builtin_nontemporal_load(&cp[t]);     // TH=NT: single-use stream
        s += x.x + x.y + x.z + x.w;
        ss = fmaf(x.x, x.x, fmaf(x.y, x.y, fmaf(x.z, x.z, fmaf(x.w, x.w, ss))));
        if (t < SV4) {
            v4f c4 = prevctx[CV4 + t];                  // hot in L2 across blocks
            dt = fmaf(x.x, c4.x, fmaf(x.y, c4.y, fmaf(x.z, c4.z, fmaf(x.w, c4.w, dt))));
            cs = fmaf(c4.x, c4.x, fmaf(c4.y, c4.y, fmaf(c4.z, c4.z, fmaf(c4.w, c4.w, cs))));
            sts = ss;   // squares accumulated so far == exactly the start region's
        }
    }
#pragma unroll
    for (int i = 1; i < ITERS; ++i) {
        v4f x = __builtin_nontemporal_load(&cp[t + i * TPB]);
        s += x.x + x.y + x.z + x.w;
        ss = fmaf(x.x, x.x, fmaf(x.y, x.y, fmaf(x.z, x.z, fmaf(x.w, x.w, ss))));
    }

    // wave32 tree reduction of the 5 partials; one slot per wave into LDS.
    float vals[5] = {s, ss, dt, sts, cs};
    const int lane = t & 31;
    const int wid  = t >> 5;
#pragma unroll
    for (int q = 0; q < 5; ++q) {
        float v = vals[q];
#pragma unroll
        for (int o = 16; o > 0; o >>= 1) v += __shfl_down(v, o, 32);
        if (lane == 0) smem[q * 8 + wid] = v;
    }
    __syncthreads();

    // Final cross-wave reduction via two chained V_WMMA_F32_16X16X4_F32 ops.
    // A (16x4 f32 layout): lanes 0-15 hold M=lane with K0/K1 in the two VGPRs,
    // lanes 16-31 hold M=lane-16 with K2/K3. We place partial[q][wave k] at
    // A[m=q][k]; B = all-ones so D[m][n] = sum_k A[m][k] (layout-robust).
    // Two WMMAs (waves 0-3 then 4-7) accumulate through C.
    // D layout: VGPR r, lanes 0-15 = row M=r -> lane 0 reads acc[0..4].
    if (t < 32) {               // wave-uniform: wave 0 has EXEC all-ones (WMMA req)
        const int  m  = t & 15;
        const bool hi = t >= 16;
        v2f a0 = {0.f, 0.f}, a1 = {0.f, 0.f};
        if (m < 5) {
            const float* r = &smem[m * 8];
            const int k = hi ? 2 : 0;
            a0.x = r[k];     a0.y = r[k + 1];
            a1.x = r[k + 4]; a1.y = r[k + 5];
        }
        v2f b = {1.f, 1.f};
        v8f acc = {};
        acc = __builtin_amdgcn_wmma_f32_16x16x4_f32(false, a0, false, b,
                                                    (short)0, acc, false, false);
        acc = __builtin_amdgcn_wmma_f32_16x16x4_f32(false, a1, false, b,
                                                    (short)0, acc, false, false);
        if (t == 0) {
            const float sum = acc[0], sumsq = acc[1], dot = acc[2],
                        startsq = acc[3], ctxsq = acc[4];
            const float Nf  = (float)ELEMS;
            float var = (sumsq - sum * sum / Nf) / (Nf - 1.f);   // ddof=1
            float sd  = sqrtf(fmaxf(var, 0.f));
            float realism = (sd < 0.01f) ? sd * 10.f
                          : (sd > 0.5f)  ? 0.5f / sd
                                         : 1.f - fabsf(sd - 0.1f);
            float denom = fmaxf(sqrtf(startsq) * sqrtf(ctxsq), 1e-8f);
            out[n] = realism + 0.15f + 0.2f * (dot / denom);
        }
    }
}

extern "C" void kernel_launch(void* const* d_in, const int* in_sizes, int n_in,
                              void* d_out, int out_size, void* d_ws, size_t ws_size,
                              hipStream_t stream) {
    const v4f* chunks  = (const v4f*)d_in[0];
    // d_in[1] = regime_probs: unused (reference regime_consistency is constant 0.5)
    const v4f* prevctx = (const v4f*)d_in[2];
    float* out = (float*)d_out;
    const int N = in_sizes[0] / ELEMS;   // 4096 chunks
    chunkrank_kernel<<<N, TPB, 0, stream>>>(chunks, prevctx, out);
}